// CostVolume_87119116632522
// MI455X (gfx1250) — compile-verified
//
#include <hip/hip_runtime.h>

typedef __attribute__((ext_vector_type(2))) float v2f;
typedef __attribute__((ext_vector_type(8))) float v8f;

#define B_ 8
#define C_ 128
#define H_ 96
#define W_ 320
#define R_ 9          // 2*4+1 displacements per axis
#define PAD_ 4
#define KC 16         // channels staged per LDS chunk
#define KP (KC / 2)   // channel pairs per chunk
#define WTILE 160     // output columns per workgroup (half of W)
#define NT 10         // 16-wide M tiles per workgroup
#define X2W (WTILE + 16)  // x2 LDS width: covers jbase-4 .. jbase+171
#define NWAVE 30      // NT m-tiles x 3 dy-groups
#define NTHREADS (NWAVE * 32)

// LDS (floats): x1 pairs [KP][WTILE][2] = 2560, x2 pairs [R_][KP][X2W][2] = 25344
#define X1S_FLOATS (KP * WTILE * 2)
#define X2S_PER_DY (KP * X2W * 2)
#define SMEM_FLOATS (X1S_FLOATS + R_ * X2S_PER_DY)   // 27904 floats = 111616 B

// Per-lane async copy: global (saddr + 32-bit byte offset) -> LDS byte address.
// Tracked by ASYNCcnt; LDS write bypasses VGPRs entirely.
__device__ __forceinline__ void async_g2l_b32(const float* base, unsigned goff_bytes,
                                              float* lds_ptr) {
    unsigned lds_off = (unsigned)(unsigned long long)lds_ptr;  // flat addr[31:0] = LDS offset
    asm volatile("global_load_async_to_lds_b32 %0, %1, %2 offset:0"
                 :: "v"(lds_off), "v"(goff_bytes), "s"(base) : "memory");
}

__global__ __launch_bounds__(NTHREADS)
void cost_volume_wmma(const float* __restrict__ x1,
                      const float* __restrict__ x2,
                      float* __restrict__ out)
{
    __shared__ float smem[SMEM_FLOATS];
    float* x1s = smem;                   // [KP][WTILE][2] channel-pair interleaved
    float* x2s = smem + X1S_FLOATS;      // [R_][KP][X2W][2]

    const int tid  = threadIdx.x;
    const int lane = tid & 31;
    const int wave = tid >> 5;
    const int mtile   = wave % NT;
    const int dygroup = wave / NT;       // 0..2 -> dy in {3g, 3g+1, 3g+2}

    const int half  = blockIdx.x;
    const int i     = blockIdx.y;        // image row
    const int b     = blockIdx.z;        // batch
    const int jbase = half * WTILE;
    const int j0    = jbase + mtile * 16;

    const int hi   = lane >> 4;          // 0: lanes 0-15, 1: lanes 16-31
    const int msub = lane & 15;
    const int col  = mtile * 16 + msub;  // LDS column this lane owns

    v8f zero = {0.f, 0.f, 0.f, 0.f, 0.f, 0.f, 0.f, 0.f};
    v8f acc1[3] = {zero, zero, zero};    // N = j0-4 .. j0+11
    v8f acc2[3] = {zero, zero, zero};    // N = j0+12 .. j0+27

    for (int c0 = 0; c0 < C_; c0 += KC) {
        const bool pf = (c0 + KC) < C_;  // prefetch next chunk into L2

        // ---- stage x1 row chunk (async global -> LDS, pair-interleaved) ----
        for (int idx = tid; idx < KC * WTILE; idx += NTHREADS) {
            int k = idx / WTILE, j = idx - k * WTILE;
            unsigned goff =
                (unsigned)(((b * C_ + (c0 + k)) * H_ + i) * W_ + (jbase + j)) * 4u;
            async_g2l_b32(x1, goff, &x1s[(k >> 1) * (WTILE * 2) + j * 2 + (k & 1)]);
            if (pf && (tid & 31) == 0)
                __builtin_prefetch(x1 + goff / 4u + KC * H_ * W_, 0, 0);
        }
        // ---- stage 9 shifted x2 rows (async, zero-padded halo via ds_store) ----
        #pragma unroll 1
        for (int dy = 0; dy < R_; ++dy) {
            int row = i + dy - PAD_;
            float* dst = x2s + dy * X2S_PER_DY;
            if (row >= 0 && row < H_) {
                for (int idx = tid; idx < KC * X2W; idx += NTHREADS) {
                    int k = idx / X2W, jj = idx - k * X2W;
                    int jg = jbase - PAD_ + jj;
                    float* ldst = &dst[(k >> 1) * (X2W * 2) + jj * 2 + (k & 1)];
                    if (jg >= 0 && jg < W_) {
                        unsigned goff =
                            (unsigned)(((b * C_ + (c0 + k)) * H_ + row) * W_ + jg) * 4u;
                        async_g2l_b32(x2, goff, ldst);
                        if (pf && (tid & 31) == 0)
                            __builtin_prefetch(x2 + goff / 4u + KC * H_ * W_, 0, 0);
                    } else {
                        *ldst = 0.f;
                    }
                }
            } else {
                for (int idx = tid; idx < KC * X2W; idx += NTHREADS)
                    dst[idx] = 0.f;
            }
        }
        asm volatile("s_wait_asynccnt 0" ::: "memory");  // LDS writes landed
        __syncthreads();

        // ---- banded GEMM over this K chunk: fragments are single b64 loads ----
        #pragma unroll
        for (int kk = 0; kk < KC; kk += 4) {
            int p = (kk >> 1) + hi;      // channel-pair index for this lane's VGPR pair
            v2f a  = *(const v2f*)&x1s[(p * WTILE + col) * 2];
            #pragma unroll
            for (int dyi = 0; dyi < 3; ++dyi) {
                int dy = dygroup * 3 + dyi;
                const float* basep = &x2s[dy * X2S_PER_DY + (p * X2W + col) * 2];
                v2f b1 = *(const v2f*)(basep);        // N = j0-4 .. j0+11
                v2f b2 = *(const v2f*)(basep + 32);   // N = j0+12 .. j0+27
                acc1[dyi] = __builtin_amdgcn_wmma_f32_16x16x4_f32(
                    false, a, false, b1, (short)0, acc1[dyi], false, false);
                acc2[dyi] = __builtin_amdgcn_wmma_f32_16x16x4_f32(
                    false, a, false, b2, (short)0, acc2[dyi], false, false);
            }
        }
        __syncthreads();
    }

    // ---- band extraction: D1[m][m+dx] (m+dx<16) / D2[m][m+dx-16] ----
    float* stage = smem + wave * 512;    // compute LDS is dead after final barrier
    const float inv_c = 1.0f / (float)C_;
    #pragma unroll
    for (int dyi = 0; dyi < 3; ++dyi) {
        int dy = dygroup * 3 + dyi;
        #pragma unroll
        for (int v = 0; v < 8; ++v) {
            int m = v + hi * 8;          // C/D layout: VGPR v, high lanes -> M = v + 8
            stage[m * 32 + msub]      = acc1[dyi][v];
            stage[m * 32 + 16 + msub] = acc2[dyi][v];
        }
        asm volatile("s_wait_dscnt 0" ::: "memory");
        #pragma unroll
        for (int dxp = 0; dxp < 5; ++dxp) {
            int dx = dxp * 2 + hi;
            if (dx < R_) {
                int m = msub;
                float val = stage[m * 32 + m + dx] * inv_c;
                val = val > 0.f ? val : 0.1f * val;   // leaky_relu(0.1)
                int d = dy * R_ + dx;
                out[((b * (R_ * R_) + d) * H_ + i) * W_ + j0 + m] = val;
            }
        }
        asm volatile("s_wait_dscnt 0" ::: "memory");
    }
}

extern "C" void kernel_launch(void* const* d_in, const int* in_sizes, int n_in,
                              void* d_out, int out_size, void* d_ws, size_t ws_size,
                              hipStream_t stream) {
    (void)in_sizes; (void)n_in; (void)out_size; (void)d_ws; (void)ws_size;
    const float* x1 = (const float*)d_in[0];
    const float* x2 = (const float*)d_in[1];
    float* out = (float*)d_out;
    dim3 grid(2, H_, B_);
    dim3 block(NTHREADS);
    hipLaunchKernelGGL(cost_volume_wmma, grid, block, 0, stream, x1, x2, out);
}